// GpuCvxpyLayer_28913719836698
// MI455X (gfx1250) — compile-verified
//
#include <hip/hip_runtime.h>
#include <math.h>

#define BB 64
#define PT 2048
#define NN 256
#define MM 128
#define MN (MM*NN)

typedef float v2f __attribute__((ext_vector_type(2)));
typedef float v8f __attribute__((ext_vector_type(8)));

// ---- workspace layout (float offsets) ----
#define OFS_PT    0                                  // pT  [PT][BB]
#define OFS_ZP    (OFS_PT + (size_t)PT*BB)           // z=CSR_P(p)  [NN][BB]
#define OFS_QACC  (OFS_ZP + (size_t)NN*BB)           // q   [NN][BB]
#define OFS_BACC  (OFS_QACC + (size_t)NN*BB)         // b   [MM][BB]
#define OFS_AACC  (OFS_BACC + (size_t)MM*BB)         // A   [MN][BB]
#define OFS_DINV  (OFS_AACC + (size_t)MN*BB)
#define OFS_RD    (OFS_DINV + (size_t)NN*BB)         // sqrt(1/Pdiag)
#define OFS_QS    (OFS_RD + (size_t)NN*BB)           // Dinv*q
#define OFS_QH    (OFS_QS + (size_t)NN*BB)           // rD*q
#define OFS_AHAT  (OFS_QH + (size_t)NN*BB)           // Ahat [BB][MM][NN]
#define OFS_S     OFS_AACC                           // alias (Aacc dead after Ahat build)
#define OFS_RR    (OFS_S + (size_t)BB*MM*MM)         // rhs / y  [BB][MM]

__global__ void k_zero(float* __restrict__ p, long long n) {
    long long i = blockIdx.x * (long long)blockDim.x + threadIdx.x;
    if (i < n) p[i] = 0.0f;
}

__global__ void k_transpose_p(const float* __restrict__ p, float* __restrict__ pT) {
    int i = blockIdx.x * blockDim.x + threadIdx.x;   // i = c*64 + b
    if (i >= PT * BB) return;
    int c = i >> 6, b = i & 63;
    pT[i] = p[b * PT + c];
}

// one thread per (nnz, batch); 64 consecutive lanes share one nnz -> coalesced p & atomics
__global__ void k_scatter(const float* __restrict__ vals, const int* __restrict__ cols,
                          const int* __restrict__ rows, const float* __restrict__ pT,
                          float* __restrict__ acc, int nnz) {
    long long t = blockIdx.x * (long long)blockDim.x + threadIdx.x;
    if (t >= (long long)nnz * BB) return;
    int k = (int)(t >> 6);
    int b = (int)(t & 63);
    int c = cols[k]; int r = rows[k]; float v = vals[k];
    atomicAdd(acc + (long long)r * BB + b, v * pT[(long long)c * BB + b]);
}

__global__ void k_prep(const float* __restrict__ zP, const float* __restrict__ qacc,
                       float* __restrict__ Dinv, float* __restrict__ rD,
                       float* __restrict__ qs, float* __restrict__ qh) {
    int i = blockIdx.x * blockDim.x + threadIdx.x;
    if (i >= NN * BB) return;
    float z = zP[i];
    float d = z * z + 1.0f;           // P_diag >= 1
    float di = 1.0f / d;
    float rd = sqrtf(di);
    float q = qacc[i];
    Dinv[i] = di; rD[i] = rd; qs[i] = di * q; qh[i] = rd * q;
}

// transpose Aacc[MN][BB] -> Ahat[b][m][n] = A * rD[n]   (32x32 LDS tiles)
__global__ void k_build_ahat(const float* __restrict__ Aacc, const float* __restrict__ rD,
                             float* __restrict__ Ahat) {
    __shared__ float tile[32][33];
    int tx = threadIdx.x, ty = threadIdx.y;      // (32,8)
    int ra0 = blockIdx.x * 32;                   // row index in [0, MN)
    int b0  = blockIdx.y * 32;                   // batch
#pragma unroll
    for (int j = 0; j < 4; ++j)
        tile[ty + 8 * j][tx] = Aacc[(long long)(ra0 + ty + 8 * j) * BB + (b0 + tx)];
    __syncthreads();
#pragma unroll
    for (int j = 0; j < 4; ++j) {
        int ra = ra0 + tx;
        int b  = b0 + ty + 8 * j;
        int n  = ra & (NN - 1);
        Ahat[(size_t)b * MN + ra] = tile[tx][ty + 8 * j] * rD[n * BB + b];
    }
}

// S[b] = Ahat[b] * Ahat[b]^T via V_WMMA_F32_16X16X4_F32.
// Block: 8 waves -> 32x64 region of S. Grid: (BB, 128/32, 128/64).
__global__ void __launch_bounds__(256, 1)
k_schur_gemm(const float* __restrict__ Ahat, float* __restrict__ S) {
    __shared__ float As[32 * 65];
    __shared__ float Bs[64 * 65];
    const int b    = blockIdx.x;
    const int i0   = blockIdx.y * 32;
    const int j0   = blockIdx.z * 64;
    const int tid  = threadIdx.x;
    const int lane = tid & 31;
    const int wv   = tid >> 5;       // 0..7
    const int tr   = wv >> 2;        // 0..1 (16-row tile)
    const int tc   = wv & 3;         // 0..3 (16-col tile)
    const int m    = lane & 15;
    const int half = lane >> 4;
    const int koff = half * 2;       // lanes 16..31 hold K=2,3 of each step
    const float* Ab = Ahat + (size_t)b * MN;
    v8f acc = {};

    for (int p = 0; p < 4; ++p) {          // K panels of 64 (total K = 256)
        const int k0 = p * 64;
        for (int idx = tid; idx < 32 * 64; idx += 256) {
            int r = idx >> 6, c = idx & 63;
            As[r * 65 + c] = Ab[(i0 + r) * NN + k0 + c];
        }
        for (int idx = tid; idx < 64 * 64; idx += 256) {
            int r = idx >> 6, c = idx & 63;
            Bs[r * 65 + c] = Ab[(j0 + r) * NN + k0 + c];
        }
        __syncthreads();
        const float* Ar = As + (tr * 16 + m) * 65 + koff;
        const float* Br = Bs + (tc * 16 + m) * 65 + koff;
#pragma unroll
        for (int kk = 0; kk < 64; kk += 4) {
            v2f a, bv;
            a.x  = Ar[kk]; a.y  = Ar[kk + 1];
            bv.x = Br[kk]; bv.y = Br[kk + 1];
            acc = __builtin_amdgcn_wmma_f32_16x16x4_f32(
                false, a, false, bv, (short)0, acc, false, false);
        }
        __syncthreads();
    }
    // C/D layout: lane n=lane&15 is column; VGPR r -> row r + 8*half
    int col  = j0 + tc * 16 + (lane & 15);
    int row0 = i0 + tr * 16 + half * 8;
    float* Sb = S + (size_t)b * MM * MM;
#pragma unroll
    for (int r = 0; r < 8; ++r)
        Sb[(row0 + r) * MM + col] = acc[r];
}

// rr[b][m] = -(b[m] + sum_n Ahat[b][m][n] * qh[n])     (= -(b + A D^-1 q))
__global__ void k_rhs(const float* __restrict__ Ahat, const float* __restrict__ qh,
                      const float* __restrict__ bacc, float* __restrict__ rr) {
    int t = blockIdx.x * blockDim.x + threadIdx.x;
    if (t >= BB * MM) return;
    int b = t >> 7, mrow = t & 127;
    const float* row = Ahat + (size_t)b * MN + mrow * NN;
    float s = 0.0f;
#pragma unroll 4
    for (int n = 0; n < NN; ++n) s += row[n] * qh[n * BB + b];
    rr[t] = -(bacc[mrow * BB + b] + s);
}

// Cholesky of S (SPD 128x128, in 64KB LDS) + L z = r, L^T y = z. One WG / batch.
__global__ void __launch_bounds__(128, 1)
k_chol_solve(const float* __restrict__ Sg, float* __restrict__ rr_y,
             float* __restrict__ dual_out) {
    __shared__ float s[MM * MM];                 // exactly 64 KB
    const int b = blockIdx.x, tid = threadIdx.x;
    for (int i = tid; i < MM * MM; i += 128) s[i] = Sg[(size_t)b * MM * MM + i];
    __syncthreads();
    // right-looking Cholesky (lower triangle only)
    for (int k = 0; k < MM; ++k) {
        if (tid == k) s[k * MM + k] = sqrtf(s[k * MM + k]);
        __syncthreads();
        float lkk = s[k * MM + k];
        if (tid > k) s[tid * MM + k] /= lkk;
        __syncthreads();
        if (tid > k) {
            float lik = s[tid * MM + k];
            for (int j = k + 1; j <= tid; ++j) s[tid * MM + j] -= lik * s[j * MM + k];
        }
        __syncthreads();
    }
    // broadcast slot: element (0,1) — strictly upper, never touched by lower-tri algo
#define BCAST s[1]
    float zi = rr_y[b * MM + tid];
    for (int k = 0; k < MM; ++k) {               // forward: L z = r
        if (tid == k) BCAST = zi / s[k * MM + k];
        __syncthreads();
        float zk = BCAST;
        if (tid > k) zi -= s[tid * MM + k] * zk;
        else if (tid == k) zi = zk;
        __syncthreads();
    }
    float yi = zi;
    for (int k = MM - 1; k >= 0; --k) {          // backward: L^T y = z
        if (tid == k) BCAST = yi / s[k * MM + k];
        __syncthreads();
        float yk = BCAST;
        if (tid < k) yi -= s[k * MM + tid] * yk;
        else if (tid == k) yi = yk;
        __syncthreads();
    }
#undef BCAST
    rr_y[b * MM + tid] = yi;                     // y (reuse rr storage)
    dual_out[b * MM + tid] = yi;
}

// x[n] = -(Dinv*q[n] + rD[n] * sum_m Ahat[b][m][n]*y[m]);   one WG / batch, n = tid
__global__ void k_primal(const float* __restrict__ Ahat, const float* __restrict__ yws,
                         const float* __restrict__ qs, const float* __restrict__ rD,
                         float* __restrict__ out) {
    __shared__ float ysh[MM];
    const int b = blockIdx.x, n = threadIdx.x;   // 256 threads
    if (n < MM) ysh[n] = yws[b * MM + n];
    __syncthreads();
    const float* Ab = Ahat + (size_t)b * MN;
    float g = 0.0f;
#pragma unroll 4
    for (int m = 0; m < MM; ++m) g += Ab[m * NN + n] * ysh[m];
    out[b * NN + n] = -(qs[n * BB + b] + rD[n * BB + b] * g);
}

extern "C" void kernel_launch(void* const* d_in, const int* in_sizes, int n_in,
                              void* d_out, int out_size, void* d_ws, size_t ws_size,
                              hipStream_t stream) {
    const float* p_stack = (const float*)d_in[0];
    const float* P_vals  = (const float*)d_in[1];
    const int*   P_cols  = (const int*)d_in[2];
    const int*   P_rows  = (const int*)d_in[3];
    const float* q_vals  = (const float*)d_in[4];
    const int*   q_cols  = (const int*)d_in[5];
    const int*   q_rows  = (const int*)d_in[6];
    const float* A_vals  = (const float*)d_in[7];
    const int*   A_cols  = (const int*)d_in[8];
    const int*   A_rows  = (const int*)d_in[9];
    const float* b_vals  = (const float*)d_in[10];
    const int*   b_cols  = (const int*)d_in[11];
    const int*   b_rows  = (const int*)d_in[12];

    float* ws   = (float*)d_ws;
    float* pT   = ws + OFS_PT;
    float* zP   = ws + OFS_ZP;
    float* qacc = ws + OFS_QACC;
    float* bacc = ws + OFS_BACC;
    float* Aacc = ws + OFS_AACC;
    float* Dinv = ws + OFS_DINV;
    float* rD   = ws + OFS_RD;
    float* qs   = ws + OFS_QS;
    float* qh   = ws + OFS_QH;
    float* Ahat = ws + OFS_AHAT;
    float* Sg   = ws + OFS_S;      // aliases Aacc (dead by then)
    float* rr   = ws + OFS_RR;
    float* out  = (float*)d_out;

    // zero all scatter accumulators (zP, qacc, bacc, Aacc are contiguous)
    long long nz = (long long)NN * BB * 2 + (long long)MM * BB + (long long)MN * BB;
    k_zero<<<(unsigned)((nz + 255) / 256), 256, 0, stream>>>(zP, nz);
    k_transpose_p<<<(PT * BB + 255) / 256, 256, 0, stream>>>(p_stack, pT);

    {   // COO scatters (batched matvec contributions)
        long long t;
        t = (long long)in_sizes[1] * BB;
        k_scatter<<<(unsigned)((t + 255) / 256), 256, 0, stream>>>(P_vals, P_cols, P_rows, pT, zP, in_sizes[1]);
        t = (long long)in_sizes[4] * BB;
        k_scatter<<<(unsigned)((t + 255) / 256), 256, 0, stream>>>(q_vals, q_cols, q_rows, pT, qacc, in_sizes[4]);
        t = (long long)in_sizes[7] * BB;
        k_scatter<<<(unsigned)((t + 255) / 256), 256, 0, stream>>>(A_vals, A_cols, A_rows, pT, Aacc, in_sizes[7]);
        t = (long long)in_sizes[10] * BB;
        k_scatter<<<(unsigned)((t + 255) / 256), 256, 0, stream>>>(b_vals, b_cols, b_rows, pT, bacc, in_sizes[10]);
    }

    k_prep<<<(NN * BB + 255) / 256, 256, 0, stream>>>(zP, qacc, Dinv, rD, qs, qh);
    k_build_ahat<<<dim3(MN / 32, BB / 32), dim3(32, 8), 0, stream>>>(Aacc, rD, Ahat);
    k_schur_gemm<<<dim3(BB, MM / 32, MM / 64), 256, 0, stream>>>(Ahat, Sg);
    k_rhs<<<(BB * MM + 255) / 256, 256, 0, stream>>>(Ahat, qh, bacc, rr);
    k_chol_solve<<<BB, MM, 0, stream>>>(Sg, rr, out + (size_t)BB * NN);  // dual
    k_primal<<<BB, NN, 0, stream>>>(Ahat, rr, qs, rD, out);              // primal
}